// MCPBRNN_Generic_Scaling_10196252360841
// MI455X (gfx1250) — compile-verified
//
#include <hip/hip_runtime.h>
#include <hip/hip_bf16.h>
#include <math.h>

#define SPIN_LEN 1000
#define TRAIN_LEN 1000000
#define ML_C 2.9086f
#define SL_C 1.898f

typedef __attribute__((ext_vector_type(2))) float v2f;
typedef __attribute__((ext_vector_type(8))) float v8f;

__device__ __forceinline__ float fast_sigmoid(float t) {
    // sigmoid(t) = 1/(1+exp(-t)) = 1/(1+exp2(-t*log2 e)) ; v_exp_f32 + v_rcp_f32
    float e = __builtin_amdgcn_exp2f(-1.44269504f * t);
    return __builtin_amdgcn_rcpf(1.0f + e);
}

// ---------------------------------------------------------------------------
// Kernel 1: sum & sum-of-squares of y_obs[SPIN_LEN:TRAIN_LEN] via WMMA.
// B matrix = all ones => D[m][n] = sum_k A[m][k] + C[m][n]; total recovered as
// (sum over all 32 lanes of the 8 accumulator VGPRs) / 16.
// Loop bounds are forced into SGPRs (readfirstlane) so the loop is scalar and
// EXEC is provably all-ones at every WMMA (ISA requirement).
// ---------------------------------------------------------------------------
__global__ void k_reduce_std(const float* __restrict__ y, int count,
                             double* __restrict__ acc /* acc[0]=sum acc[1]=sumsq */) {
    const int lane = threadIdx.x & 31;
    // wave-uniform values -> force into SGPRs so loop control is SALU-only
    const int wave   = __builtin_amdgcn_readfirstlane(
        (int)((blockIdx.x * blockDim.x + threadIdx.x) >> 5));
    const int nwaves = __builtin_amdgcn_readfirstlane(
        (int)((gridDim.x * blockDim.x) >> 5));
    const int scount = __builtin_amdgcn_readfirstlane(count);

    float s, q;
#if __has_builtin(__builtin_amdgcn_wmma_f32_16x16x4_f32)
    v8f cs = {}; v8f cq = {};
    v2f ones; ones[0] = 1.0f; ones[1] = 1.0f;
    for (int base = wave * 64; base < scount; base += nwaves * 64) {
        int i0 = base + lane * 2;
        int i1 = i0 + 1;
        int m  = scount - 1;
        float a0 = y[i0 < scount ? i0 : m];
        float a1 = y[i1 < scount ? i1 : m];
        a0 = (i0 < scount) ? a0 : 0.0f;   // v_cndmask select, no EXEC divergence
        a1 = (i1 < scount) ? a1 : 0.0f;
        v2f A; A[0] = a0;      A[1] = a1;
        v2f Q; Q[0] = a0 * a0; Q[1] = a1 * a1;
        cs = __builtin_amdgcn_wmma_f32_16x16x4_f32(false, A, false, ones,
                                                   (short)0, cs, false, false);
        cq = __builtin_amdgcn_wmma_f32_16x16x4_f32(false, Q, false, ones,
                                                   (short)0, cq, false, false);
    }
    s = cs[0]+cs[1]+cs[2]+cs[3]+cs[4]+cs[5]+cs[6]+cs[7];
    q = cq[0]+cq[1]+cq[2]+cq[3]+cq[4]+cq[5]+cq[6]+cq[7];
    // every column of D holds identical row-sums -> 32-lane sum counts total 16x
    for (int off = 16; off; off >>= 1) {
        s += __shfl_xor(s, off, 32);
        q += __shfl_xor(q, off, 32);
    }
    s *= (1.0f / 16.0f);
    q *= (1.0f / 16.0f);
#else
    s = 0.0f; q = 0.0f;
    for (int base = wave * 64; base < scount; base += nwaves * 64) {
        for (int k = 0; k < 2; ++k) {
            int i = base + lane * 2 + k;
            if (i < scount) { float v = y[i]; s += v; q += v * v; }
        }
    }
    for (int off = 16; off; off >>= 1) {
        s += __shfl_xor(s, off, 32);
        q += __shfl_xor(q, off, 32);
    }
#endif
    if (lane == 0) {
        atomicAdd(&acc[0], (double)s);   // global_atomic_add_f64
        atomicAdd(&acc[1], (double)q);
    }
}

// ---------------------------------------------------------------------------
// Kernel 2: parallel precompute of ol[b] (depends only on u2) straight into
// the gate_ol output slot; zero the padded prefix of c_n / gate_oo so the
// serial scan never touches b < time_lag.
// ---------------------------------------------------------------------------
__global__ void k_gates(const float2* __restrict__ x2,
                        float* __restrict__ gate_ol_out,
                        float* __restrict__ c_out,
                        float* __restrict__ oo_out,
                        const float* __restrict__ wr_o,
                        const float* __restrict__ wr_l,
                        const float* __restrict__ wr_f,
                        const float* __restrict__ b0l_p,
                        const float* __restrict__ wb2_p,
                        const int*  __restrict__ tl_p,
                        int B) {
    int b = blockIdx.x * blockDim.x + threadIdx.x;
    if (b >= B) return;
    float eo  = expf(wr_o[0]);
    float el  = expf(wr_l[0]);
    float ef  = expf(wr_f[0]);
    float ol1 = el / (eo + el + ef);
    float wb2 = wb2_p[0] * (1.0f / SL_C);
    float b0l = b0l_p[0];
    float u2  = x2[b].y;
    float ol  = ol1 * fast_sigmoid(fmaf(u2 - ML_C, wb2, b0l));
    int tl = tl_p[0];
    if (b < tl) {
        gate_ol_out[b] = 0.0f;
        c_out[b]       = 0.0f;
        oo_out[b]      = 0.0f;
    } else {
        gate_ol_out[b] = ol;
    }
}

// ---------------------------------------------------------------------------
// Kernel 3: the unavoidable sequential scan over B. Single lane carries c.
// Critical chain per step: fma -> v_exp_f32 -> add -> v_rcp_f32 -> fma.
// Tiled by 16 so the prefetch sits outside the hot body and the inner loop
// has a fixed trip count (compiler can unroll + batch the off-chain loads).
// ---------------------------------------------------------------------------
__global__ void k_scan(const float2* __restrict__ x2,
                       const float* __restrict__ gate_ol,
                       float* __restrict__ c_out,
                       float* __restrict__ oo_out,
                       const float* __restrict__ cmean,
                       const float* __restrict__ cstd,
                       const float* __restrict__ wr_o,
                       const float* __restrict__ wr_l,
                       const float* __restrict__ wr_f,
                       const float* __restrict__ b0o_p,
                       const float* __restrict__ wb1_p,
                       const int*  __restrict__ tl_p,
                       int B) {
    float mo  = cmean[0];
    float so  = cstd[0];
    float eo  = expf(wr_o[0]);
    float el  = expf(wr_l[0]);
    float ef  = expf(wr_f[0]);
    float oo1 = eo / (eo + el + ef);
    float beta  = wb1_p[0] / so;
    float alpha = b0o_p[0] - mo * beta;
    // sigmoid(alpha + beta*c) = rcp(1 + exp2(a2 + b2*c))
    const float L2E = 1.44269504f;
    float a2 = -alpha * L2E;
    float b2 = -beta  * L2E;
    int tl = tl_p[0];

    float c = 0.0f;

    #define SCAN_STEP(b)                                                    \
        {                                                                   \
            float u1 = x2[(b)].x;                                           \
            float ol = gate_ol[(b)];                                        \
            /* dependent chain */                                           \
            float t  = __builtin_fmaf(b2, c, a2);                           \
            float e  = __builtin_amdgcn_exp2f(t);                           \
            float r  = __builtin_amdgcn_rcpf(1.0f + e);                     \
            /* off-chain (overlaps the trans ops) */                        \
            float oc = oo1 * c;                                             \
            float cn = __builtin_fmaf(1.0f - ol, c, u1);                    \
            c_out[(b)]  = c;          /* carry BEFORE update */             \
            oo_out[(b)] = oo1 * r;                                          \
            c = __builtin_fmaf(-oc, r, cn); /* (1-oo-ol)*c + u1 */          \
        }

    int b = tl;
    // head: align to a multiple of 16
    for (; b < B && (b & 15); ++b) SCAN_STEP(b);
    // main tiles of 16 with prefetch ~256 elements ahead
    for (; b + 16 <= B; b += 16) {
        __builtin_prefetch(&x2[b + 256], 0, 3);          // global_prefetch_b8
        if ((b & 31) == 0)
            __builtin_prefetch(&gate_ol[b + 256], 0, 3); // global_prefetch_b8
        #pragma unroll
        for (int k = 0; k < 16; ++k) SCAN_STEP(b + k);
    }
    // tail
    for (; b < B; ++b) SCAN_STEP(b);
    #undef SCAN_STEP
}

// ---------------------------------------------------------------------------
// Kernel 4: elementwise finalize: h_n, l_n, bp_n, gate_ib, gate_f, h_nout,
// obs_std from (c0, oo, ol) and the reduction result.
// ---------------------------------------------------------------------------
__global__ void k_finalize(float* __restrict__ out,
                           const double* __restrict__ acc,
                           const int* __restrict__ tl_p,
                           int B, int n) {
    int b = blockIdx.x * blockDim.x + threadIdx.x;
    if (b >= B) return;
    const float* c_n  = out + (size_t)B;
    const float* g_oo = out + (size_t)5 * B;
    const float* g_ol = out + (size_t)6 * B;
    float c0 = c_n[b];
    float oo = g_oo[b];
    float ol = g_ol[b];
    double s  = acc[0];
    double q  = acc[1];
    double var = (q - s * s / (double)n) / (double)(n - 1);
    float ostd = (float)sqrt(var);
    int tl = tl_p[0];
    bool pad = (b < tl);
    float h = oo * c0;
    out[b]                         = h;                       // h_n
    out[(size_t)2 * B + b]         = ol * c0;                 // l_n
    out[(size_t)3 * B + b]         = 0.0f;                    // bp_n
    out[(size_t)4 * B + b]         = 0.0f;                    // gate_ib
    out[(size_t)7 * B + b]         = pad ? 0.0f : (1.0f - oo - ol); // gate_f
    out[(size_t)8 * B + 2 * b]     = h;                       // h_nout[:,0]
    out[(size_t)8 * B + 2 * b + 1] = pad ? 0.0f : ostd;       // h_nout[:,1]
    out[(size_t)10 * B + b]        = pad ? 0.0f : ostd;       // obs_std
}

// ---------------------------------------------------------------------------
extern "C" void kernel_launch(void* const* d_in, const int* in_sizes, int n_in,
                              void* d_out, int out_size, void* d_ws, size_t ws_size,
                              hipStream_t stream) {
    const float2* x2    = (const float2*)d_in[0];
    const int*    tl_p  = (const int*)  d_in[2];
    const float*  y_obs = (const float*)d_in[3];
    const float*  cmean = (const float*)d_in[4];
    const float*  cstd  = (const float*)d_in[5];
    const float*  wr_o  = (const float*)d_in[6];
    const float*  wr_l  = (const float*)d_in[7];
    const float*  wr_f  = (const float*)d_in[8];
    const float*  b0o   = (const float*)d_in[9];
    const float*  wb1   = (const float*)d_in[10];
    const float*  b0l   = (const float*)d_in[11];
    const float*  wb2   = (const float*)d_in[12];

    int B = in_sizes[0] / 2;   // x is (B,1,2)
    float* out = (float*)d_out;
    float* c_out  = out + (size_t)B;       // c_n
    float* oo_out = out + (size_t)5 * B;   // gate_oo
    float* ol_out = out + (size_t)6 * B;   // gate_ol

    double* acc = (double*)d_ws;
    hipMemsetAsync(acc, 0, 2 * sizeof(double), stream);

    int ylen = in_sizes[3];
    int hi   = (ylen < TRAIN_LEN) ? ylen : TRAIN_LEN;
    int count = hi - SPIN_LEN;
    if (count < 2) count = 2;

    // 1) std reduction via WMMA matrix pipe
    k_reduce_std<<<512, 256, 0, stream>>>(y_obs + SPIN_LEN, count, acc);

    // 2) parallel ol gate (input-only)
    int blk = 256;
    k_gates<<<(B + blk - 1) / blk, blk, 0, stream>>>(x2, ol_out, c_out, oo_out,
                                                     wr_o, wr_l, wr_f, b0l, wb2,
                                                     tl_p, B);

    // 3) serial nonlinear scan (dependency-chain bound)
    k_scan<<<1, 1, 0, stream>>>(x2, ol_out, c_out, oo_out, cmean, cstd,
                                wr_o, wr_l, wr_f, b0o, wb1, tl_p, B);

    // 4) elementwise finalize of all remaining outputs
    k_finalize<<<(B + blk - 1) / blk, blk, 0, stream>>>(out, acc, tl_p, B, count);
}